// GMMLoss_fast_73547019977335
// MI455X (gfx1250) — compile-verified
//
#include <hip/hip_runtime.h>
#include <hip/hip_bf16.h>

typedef __attribute__((ext_vector_type(2))) float v2f;
typedef __attribute__((ext_vector_type(8))) float v8f;

#define NCLS 10
#define KDIM 64
#define SIGMA_ISO 1.0f

// ---- workspace layout (float offsets) ----
#define WS_SXX    0            // 10*64*64 = 40960
#define WS_SSUM   40960        // 10*64 = 640
#define WS_CNT    41600        // 16 (10 used)
#define WS_ZERON  41616        // everything above must be zeroed per call
#define WS_MUF    41616        // 640
#define WS_SIGMA  42256        // 40960
#define WS_INV    83216        // 40960
#define WS_LOGDET 124176       // 16 (10 used)

// ------------------------------------------------------------------
// zero the accumulated regions (harness poisons ws with 0xAA)
// ------------------------------------------------------------------
__global__ void gmm_zero_kernel(float* ws, int n) {
    int i = blockIdx.x * blockDim.x + threadIdx.x;
    if (i < n) ws[i] = 0.0f;
}

// ------------------------------------------------------------------
// Heavy kernel: per-class second moments Sxx[c] = mu^T diag(1_c) mu
// 10 waves per workgroup, wave w owns class w, full 64x64 f32
// accumulator as 4x4 tiles of V_WMMA_F32_16X16X4_F32.
// ------------------------------------------------------------------
__global__ __launch_bounds__(320) void gmm_sxx_kernel(
    const float* __restrict__ mu, const int* __restrict__ label,
    float* __restrict__ sxx, int B) {

    __shared__ float tile[64][65];     // 64 staged rows, padded
    __shared__ int   lcls[64];
    __shared__ unsigned chunkmask[4];  // class-presence bits per 16-row sub-chunk

    const int tid  = threadIdx.x;
    const int wv   = tid >> 5;         // wave id == class id (0..9)
    const int lane = tid & 31;
    const int hi   = lane >> 4;        // 0/1: which K-half of fragment
    const int lcol = lane & 15;        // M (for A) / N (for B) within tile

    v8f acc[4][4];
#pragma unroll
    for (int mt = 0; mt < 4; ++mt)
#pragma unroll
        for (int nt = 0; nt < 4; ++nt)
            acc[mt][nt] = (v8f){0.f,0.f,0.f,0.f,0.f,0.f,0.f,0.f};

    for (int base = blockIdx.x * 64; base < B; base += gridDim.x * 64) {
        // cooperative stage: 64 rows x 64 cols
        for (int i = tid; i < 64 * 64; i += 320) {
            int r = i >> 6, c = i & 63;
            tile[r][c] = mu[(base + r) * 64 + c];
        }
        if (tid < 64) lcls[tid] = label[base + tid];
        __syncthreads();
        if (tid < 4) {
            unsigned m = 0u;
#pragma unroll
            for (int r = 0; r < 16; ++r) m |= (1u << lcls[tid * 16 + r]);
            chunkmask[tid] = m;
        }
        __syncthreads();

#pragma unroll 1
        for (int sub = 0; sub < 4; ++sub) {
            if (((chunkmask[sub] >> wv) & 1u) == 0u) continue;  // wave-uniform skip
            const int rbase = sub * 16;
#pragma unroll
            for (int kk = 0; kk < 4; ++kk) {
                const int rx = rbase + 4 * kk + 2 * hi;   // row for frag .x (K = 2*hi)
                const int ry = rx + 1;                    // row for frag .y (K = 2*hi+1)
                const float mx = (lcls[rx] == wv) ? 1.0f : 0.0f;
                const float my = (lcls[ry] == wv) ? 1.0f : 0.0f;
                v2f frag[4];
#pragma unroll
                for (int t = 0; t < 4; ++t) {
                    frag[t].x = tile[rx][t * 16 + lcol];
                    frag[t].y = tile[ry][t * 16 + lcol];
                }
#pragma unroll
                for (int mt = 0; mt < 4; ++mt) {
                    v2f am;
                    am.x = frag[mt].x * mx;
                    am.y = frag[mt].y * my;
#pragma unroll
                    for (int nt = 0; nt < 4; ++nt) {
                        acc[mt][nt] = __builtin_amdgcn_wmma_f32_16x16x4_f32(
                            false, am, false, frag[nt],
                            (short)0, acc[mt][nt], false, false);
                    }
                }
            }
        }
        __syncthreads();
    }

    // flush: C/D layout -> VGPR v holds M = v (lanes 0-15) / v+8 (lanes 16-31), N = lane%16
#pragma unroll
    for (int mt = 0; mt < 4; ++mt)
#pragma unroll
        for (int nt = 0; nt < 4; ++nt)
#pragma unroll
            for (int v = 0; v < 8; ++v) {
                int M = mt * 16 + v + 8 * hi;
                int N = nt * 16 + lcol;
                atomicAdd(&sxx[(wv * 64 + M) * 64 + N], acc[mt][nt][v]);
            }
}

// ------------------------------------------------------------------
// per-class counts + feature sums (LDS float atomics, then flush)
// ------------------------------------------------------------------
__global__ __launch_bounds__(256) void gmm_sums_kernel(
    const float* __restrict__ mu, const int* __restrict__ label,
    float* __restrict__ ssum, float* __restrict__ cnt, int B) {

    __shared__ float ls[NCLS * 64 + NCLS];  // sums then counts
    const int tid = threadIdx.x;
    for (int i = tid; i < NCLS * 64 + NCLS; i += 256) ls[i] = 0.0f;
    __syncthreads();

    const int r = tid >> 6;    // 0..3 : row slot
    const int f = tid & 63;    // feature
    for (int row = blockIdx.x * 4 + r; row < B; row += gridDim.x * 4) {
        int cls = label[row];
        atomicAdd(&ls[cls * 64 + f], mu[row * 64 + f]);
        if (f == 0) atomicAdd(&ls[NCLS * 64 + cls], 1.0f);
    }
    __syncthreads();
    for (int i = tid; i < NCLS * 64; i += 256) atomicAdd(&ssum[i], ls[i]);
    if (tid < NCLS) atomicAdd(&cnt[tid], ls[NCLS * 64 + tid]);
}

// ------------------------------------------------------------------
// per-class finalize: Sigma, Cholesky, logdet, inverse
// one block per class, 64 threads
// ------------------------------------------------------------------
__global__ __launch_bounds__(64) void gmm_finalize_kernel(
    const float* __restrict__ sxx, const float* __restrict__ ssum,
    const float* __restrict__ cnt, float* __restrict__ muF,
    float* __restrict__ sig, float* __restrict__ inv,
    float* __restrict__ logdet) {

    __shared__ float S[64][65];
    __shared__ float W[64][65];
    __shared__ float muf[64];

    const int c   = blockIdx.x;
    const int tid = threadIdx.x;
    const float n = cnt[c];
    const float rn = 1.0f / n;

    float m = ssum[c * 64 + tid] * rn;
    muf[tid] = m;
    muF[c * 64 + tid] = m;
    __syncthreads();

    for (int a = 0; a < 64; ++a) {
        float v = sxx[c * 4096 + a * 64 + tid] * rn - muf[a] * muf[tid]
                + ((a == tid) ? SIGMA_ISO : 0.0f);
        S[a][tid] = v;
        sig[c * 4096 + a * 64 + tid] = v;
    }
    __syncthreads();

    // Cholesky (lower L stored in S)
    for (int j = 0; j < 64; ++j) {
        if (tid == 0) S[j][j] = sqrtf(S[j][j]);
        __syncthreads();
        float Ljj = S[j][j];
        if (tid > j) S[tid][j] /= Ljj;
        __syncthreads();
        if (tid > j) {
            float Lij = S[tid][j];
            for (int b = j + 1; b <= tid; ++b) S[tid][b] -= Lij * S[b][j];
        }
        __syncthreads();
    }

    if (tid == 0) {
        float ld = 0.0f;
        for (int j = 0; j < 64; ++j) ld += __logf(S[j][j]);
        logdet[c] = 2.0f * ld;
    }

    // inverse: each thread solves L L^T x = e_tid, column kept in W[:][tid]
    for (int i = 0; i < 64; ++i) {
        float s = (i == tid) ? 1.0f : 0.0f;
        for (int j = 0; j < i; ++j) s -= S[i][j] * W[j][tid];
        W[i][tid] = s / S[i][i];
    }
    for (int i = 63; i >= 0; --i) {
        float s = W[i][tid];
        for (int j = i + 1; j < 64; ++j) s -= S[j][i] * W[j][tid];
        W[i][tid] = s / S[i][i];
    }
    for (int i = 0; i < 64; ++i) inv[c * 4096 + i * 64 + tid] = W[i][tid];
}

// ------------------------------------------------------------------
// pairwise KL reduction -> single scalar
// ------------------------------------------------------------------
__global__ __launch_bounds__(256) void gmm_kl_kernel(
    const float* __restrict__ sig, const float* __restrict__ inv,
    const float* __restrict__ muF, const float* __restrict__ logdet,
    float* __restrict__ out) {

    __shared__ float red[256];
    const int tid = threadIdx.x;
    float accv = 0.0f;
    for (int idx = tid; idx < NCLS * NCLS * 4096; idx += 256) {
        int p = idx >> 12;
        int e = idx & 4095;
        int i = p / NCLS, j = p - i * NCLS;
        int a = e >> 6,  b = e & 63;
        float inv_jab = inv[j * 4096 + e];                  // inv[j][a][b]
        float sig_iba = sig[i * 4096 + b * 64 + a];         // Sigma[i][b][a]
        float da = muF[i * 64 + a] - muF[j * 64 + a];
        float db = muF[i * 64 + b] - muF[j * 64 + b];
        accv += inv_jab * (sig_iba + da * db);              // tr + quad contributions
    }
    red[tid] = accv;
    __syncthreads();
    for (int s = 128; s > 0; s >>= 1) {
        if (tid < s) red[tid] += red[tid + s];
        __syncthreads();
    }
    if (tid == 0) {
        float tot = red[0];
        float extra = 0.0f;
        for (int p = 0; p < NCLS * NCLS; ++p) {
            int i = p / NCLS, j = p % NCLS;
            extra += logdet[j] - logdet[i] - (float)KDIM;
        }
        out[0] = 0.5f * (tot + extra) / (float)(NCLS * NCLS);
    }
}

// ------------------------------------------------------------------
extern "C" void kernel_launch(void* const* d_in, const int* in_sizes, int n_in,
                              void* d_out, int out_size, void* d_ws, size_t ws_size,
                              hipStream_t stream) {
    const float* mu    = (const float*)d_in[0];
    const int*   label = (const int*)d_in[1];
    float* out = (float*)d_out;
    float* ws  = (float*)d_ws;
    const int B = in_sizes[1];

    float* sxx    = ws + WS_SXX;
    float* ssum   = ws + WS_SSUM;
    float* cnt    = ws + WS_CNT;
    float* muF    = ws + WS_MUF;
    float* sig    = ws + WS_SIGMA;
    float* inv    = ws + WS_INV;
    float* logdet = ws + WS_LOGDET;

    gmm_zero_kernel<<<(WS_ZERON + 255) / 256, 256, 0, stream>>>(ws, WS_ZERON);
    gmm_sxx_kernel<<<256, 320, 0, stream>>>(mu, label, sxx, B);
    gmm_sums_kernel<<<512, 256, 0, stream>>>(mu, label, ssum, cnt, B);
    gmm_finalize_kernel<<<NCLS, 64, 0, stream>>>(sxx, ssum, cnt, muF, sig, inv, logdet);
    gmm_kl_kernel<<<1, 256, 0, stream>>>(sig, inv, muF, logdet, out);
}